// GATEncoder_77498389889099
// MI455X (gfx1250) — compile-verified
//
#include <hip/hip_runtime.h>
#include <math.h>

#define NNODES   50000
#define NEDGES   800000
#define INDIM    32
#define HID      128
#define EMB      64
#define HEADS    4
#define NB       8
#define NLAYERS  3
#define NEG_SLOPE 0.2f
#define LN_EPS    1e-5f

typedef __attribute__((ext_vector_type(2))) float v2f;
typedef __attribute__((ext_vector_type(8))) float v8f;

// ---- ordered-int encoding for float atomicMax (monotone) ----
__device__ __forceinline__ int f2ord(float f) {
    int i = __float_as_int(f);
    return (i >= 0) ? i : (i ^ 0x7fffffff);
}
__device__ __forceinline__ float ord2f(int i) {
    return __int_as_float((i >= 0) ? i : (i ^ 0x7fffffff));
}
__device__ __forceinline__ float eluf(float x) { return x > 0.f ? x : expm1f(x); }

// =====================================================================
// WMMA fp32 GEMM: Cout[M x Ncol] = A[M x K] @ W[K x Ncol] + bias
// One wave per 16x16 output tile; V_WMMA_F32_16X16X4_F32, K stepped by 4.
// A frag (16x4): lanes 0-15 -> K = k,k+1 ; lanes 16-31 -> K = k+2,k+3
// B frag (4x16): mirror layout (row striped across lanes within a VGPR).
// C/D: VGPR v holds row m0+v (lanes 0-15) / m0+v+8 (lanes 16-31), col = n0+(lane&15)
// =====================================================================
__global__ void k_gemm_wmma(const float* __restrict__ A, const float* __restrict__ W,
                            const float* __restrict__ bias, float* __restrict__ Cout,
                            int M, int K, int Ncol) {
    const int wid   = threadIdx.x >> 5;
    const int lane  = threadIdx.x & 31;
    const int tilesM = M >> 4;
    const int tilesN = Ncol >> 4;
    const int tile  = blockIdx.x * (blockDim.x >> 5) + wid;
    if (tile >= tilesM * tilesN) return;   // wave-uniform: EXEC stays all-ones for WMMA
    const int tm = tile % tilesM;
    const int tn = tile / tilesM;
    const int m0 = tm << 4, n0 = tn << 4;
    const int col   = n0 + (lane & 15);
    const int arow  = m0 + (lane & 15);
    const int khalf = (lane >> 4) << 1;    // 0 or 2

    v8f acc;
    const float bv = bias[col];
#pragma unroll
    for (int i = 0; i < 8; ++i) acc[i] = bv;

    for (int k = 0; k < K; k += 4) {
        v2f a = *(const v2f*)(A + (size_t)arow * K + (k + khalf));
        v2f b;
        b.x = W[(size_t)(k + khalf + 0) * Ncol + col];
        b.y = W[(size_t)(k + khalf + 1) * Ncol + col];
        acc = __builtin_amdgcn_wmma_f32_16x16x4_f32(false, a, false, b,
                                                    (short)0, acc, false, false);
    }

    const int rbase = m0 + ((lane >> 4) << 3);
#pragma unroll
    for (int v = 0; v < 8; ++v)
        Cout[(size_t)(rbase + v) * Ncol + col] = acc[v];
}

// ---- self-loop edge_attr = mean of incoming ----
__global__ void k_edge_count(const int* __restrict__ dst, const float* __restrict__ ea,
                             float* __restrict__ cnt, float* __restrict__ easum) {
    int e = blockIdx.x * blockDim.x + threadIdx.x;
    if (e >= NEDGES) return;
    int d = dst[e];
    atomicAdd(&cnt[d], 1.0f);
    atomicAdd(&easum[d], ea[e]);
}
__global__ void k_ea_final(float* __restrict__ easum, const float* __restrict__ cnt) {
    int i = blockIdx.x * blockDim.x + threadIdx.x;
    if (i >= NNODES) return;
    easum[i] /= fmaxf(cnt[i], 1.0f);
}

// ---- edge pass 1: alpha per head + segment max (wave per edge) ----
__global__ void k_edge_alpha(const int* __restrict__ srcArr, const int* __restrict__ dstArr,
                             const float* __restrict__ eattr, const float* __restrict__ eaself,
                             const float* __restrict__ xl, const float* __restrict__ xr,
                             const float* __restrict__ We, const float* __restrict__ att,
                             float* __restrict__ exbuf, int* __restrict__ amax) {
    int e = (blockIdx.x * blockDim.x + threadIdx.x) >> 5;
    int lane = threadIdx.x & 31;
    if (e >= NEDGES + NNODES) return;
    int s, d; float ev;
    if (e < NEDGES) { s = srcArr[e]; d = dstArr[e]; ev = eattr[e]; }
    else            { s = e - NEDGES; d = s;        ev = eaself[s]; }
    const int c0 = lane * 4;
    const float* pl = xl + (size_t)s * HID + c0;
    const float* pr = xr + (size_t)d * HID + c0;
    float partial = 0.f;
#pragma unroll
    for (int j = 0; j < 4; ++j) {
        float m = pl[j] + pr[j] + ev * We[c0 + j];
        m = (m > 0.f) ? m : NEG_SLOPE * m;
        partial += m * att[c0 + j];
    }
    // reduce within each 8-lane head group (head = lane>>3, 32 channels/head)
    partial += __shfl_xor(partial, 1, 32);
    partial += __shfl_xor(partial, 2, 32);
    partial += __shfl_xor(partial, 4, 32);
    int hh = lane >> 3;
    if ((lane & 7) == 0) {
        exbuf[(size_t)e * HEADS + hh] = partial;
        atomicMax(&amax[d * HEADS + hh], f2ord(partial));
    }
}

// ---- edge pass 2: exp + denominator (thread per edge*head) ----
__global__ void k_edge_exp(const int* __restrict__ dstArr, float* __restrict__ exbuf,
                           const int* __restrict__ amax, float* __restrict__ den) {
    int i = blockIdx.x * blockDim.x + threadIdx.x;
    if (i >= (NEDGES + NNODES) * HEADS) return;
    int e = i >> 2, hh = i & 3;
    int d = (e < NEDGES) ? dstArr[e] : (e - NEDGES);
    float ex = expf(exbuf[i] - ord2f(amax[d * HEADS + hh]));
    exbuf[i] = ex;
    atomicAdd(&den[d * HEADS + hh], ex);
}

// ---- edge pass 3: weighted scatter (wave per edge) ----
__global__ void k_edge_scatter(const int* __restrict__ srcArr, const int* __restrict__ dstArr,
                               const float* __restrict__ exbuf, const float* __restrict__ den,
                               const float* __restrict__ xl, float* __restrict__ agg) {
    int e = (blockIdx.x * blockDim.x + threadIdx.x) >> 5;
    int lane = threadIdx.x & 31;
    if (e >= NEDGES + NNODES) return;
    int s, d;
    if (e < NEDGES) { s = srcArr[e]; d = dstArr[e]; }
    else            { s = e - NEDGES; d = s; }
    int hh = lane >> 3, c0 = lane * 4;
    float a = exbuf[(size_t)e * HEADS + hh] / den[d * HEADS + hh];
    const float* pl = xl + (size_t)s * HID + c0;
    float* pa = agg + (size_t)d * HID + c0;
#pragma unroll
    for (int j = 0; j < 4; ++j) atomicAdd(&pa[j], a * pl[j]);
}

// ---- node update: elu + residual + layernorm (wave per node, wave32 shuffles) ----
__global__ void k_update_h(float* __restrict__ h, const float* __restrict__ agg,
                           const float* __restrict__ bc, const float* __restrict__ g,
                           const float* __restrict__ bta) {
    int n = (blockIdx.x * blockDim.x + threadIdx.x) >> 5;
    int lane = threadIdx.x & 31;
    if (n >= NNODES) return;
    int c0 = lane * 4;
    float t[4]; float s = 0.f, sq = 0.f;
#pragma unroll
    for (int j = 0; j < 4; ++j) {
        float o = eluf(agg[(size_t)n * HID + c0 + j] + bc[c0 + j]);
        float v = h[(size_t)n * HID + c0 + j] + o;
        t[j] = v; s += v; sq += v * v;
    }
#pragma unroll
    for (int off = 16; off >= 1; off >>= 1) {
        s  += __shfl_xor(s,  off, 32);
        sq += __shfl_xor(sq, off, 32);
    }
    float mean = s * (1.0f / HID);
    float var  = sq * (1.0f / HID) - mean * mean;
    float inv  = rsqrtf(var + LN_EPS);
#pragma unroll
    for (int j = 0; j < 4; ++j)
        h[(size_t)n * HID + c0 + j] = (t[j] - mean) * inv * g[c0 + j] + bta[c0 + j];
}

// ---- graph pooling ----
__global__ void k_batch_sum(const float* __restrict__ h, const int* __restrict__ batch,
                            float* __restrict__ gsum, float* __restrict__ gcnt) {
    int n = (blockIdx.x * blockDim.x + threadIdx.x) >> 5;
    int lane = threadIdx.x & 31;
    if (n >= NNODES) return;
    int b = batch[n];
    int c0 = lane * 4;
#pragma unroll
    for (int j = 0; j < 4; ++j)
        atomicAdd(&gsum[b * HID + c0 + j], h[(size_t)n * HID + c0 + j]);
    if (lane == 0) atomicAdd(&gcnt[b], 1.0f);
}

__global__ void k_graph_mlp(const float* __restrict__ gsum, const float* __restrict__ gcnt,
                            const float* __restrict__ Wg1, const float* __restrict__ bg1,
                            const float* __restrict__ Wg2, const float* __restrict__ bg2,
                            float* __restrict__ gout) {
    __shared__ float gm[NB * HID];
    __shared__ float g1[NB * HID];
    int t = threadIdx.x;
    for (int i = t; i < NB * HID; i += blockDim.x)
        gm[i] = gsum[i] / fmaxf(gcnt[i / HID], 1.0f);
    __syncthreads();
    for (int b = 0; b < NB; ++b) {
        float s = bg1[t];
        for (int k = 0; k < HID; ++k) s += gm[b * HID + k] * Wg1[k * HID + t];
        g1[b * HID + t] = eluf(s);
    }
    __syncthreads();
    if (t < EMB) {
        for (int b = 0; b < NB; ++b) {
            float s = bg2[t];
            for (int k = 0; k < HID; ++k) s += g1[b * HID + k] * Wg2[k * EMB + t];
            gout[b * EMB + t] = s;
        }
    }
}

extern "C" void kernel_launch(void* const* d_in, const int* in_sizes, int n_in,
                              void* d_out, int out_size, void* d_ws, size_t ws_size,
                              hipStream_t stream) {
    const float* x    = (const float*)d_in[0];
    const int*   ei   = (const int*)  d_in[1];
    const float* eatt = (const float*)d_in[2];
    const int*   batch= (const int*)  d_in[3];
    const float* W_in = (const float*)d_in[4];
    const float* b_in = (const float*)d_in[5];
    const float* Wl   = (const float*)d_in[6];
    const float* bl   = (const float*)d_in[7];
    const float* Wr   = (const float*)d_in[8];
    const float* br   = (const float*)d_in[9];
    const float* We   = (const float*)d_in[10];
    const float* att  = (const float*)d_in[11];
    const float* bc   = (const float*)d_in[12];
    const float* ln_g = (const float*)d_in[13];
    const float* ln_b = (const float*)d_in[14];
    const float* Wn   = (const float*)d_in[15];
    const float* bn   = (const float*)d_in[16];
    const float* Wg1  = (const float*)d_in[17];
    const float* bg1  = (const float*)d_in[18];
    const float* Wg2  = (const float*)d_in[19];
    const float* bg2  = (const float*)d_in[20];

    const int* srcArr = ei;
    const int* dstArr = ei + NEDGES;

    float* out       = (float*)d_out;
    float* node_emb  = out;                                   // N*EMB
    float* graph_emb = out + (size_t)NNODES * EMB;            // B*EMB
    float* h_out     = graph_emb + (size_t)NB * EMB;          // N*HID

    // ---- workspace carve ----
    float* w = (float*)d_ws;
    float* h      = w;  w += (size_t)NNODES * HID;
    float* xl     = w;  w += (size_t)NNODES * HID;
    float* xr     = w;  w += (size_t)NNODES * HID;
    float* agg    = w;  w += (size_t)NNODES * HID;
    float* eaself = w;  w += NNODES;
    float* cnt    = w;  w += NNODES;
    float* exbuf  = w;  w += (size_t)(NEDGES + NNODES) * HEADS;
    float* den    = w;  w += (size_t)NNODES * HEADS;
    int*   amax   = (int*)w; w += (size_t)NNODES * HEADS;
    float* gsum   = w;  w += NB * HID;
    float* gcnt   = w;  w += NB;

    const int etot = NEDGES + NNODES;

    // self-loop edge_attr (mean of incoming)
    hipMemsetAsync(cnt,    0, NNODES * sizeof(float), stream);
    hipMemsetAsync(eaself, 0, NNODES * sizeof(float), stream);
    k_edge_count<<<(NEDGES + 255) / 256, 256, 0, stream>>>(dstArr, eatt, cnt, eaself);
    k_ea_final<<<(NNODES + 255) / 256, 256, 0, stream>>>(eaself, cnt);

    // h = x @ W_in + b_in
    {
        int tiles = (NNODES / 16) * (HID / 16);
        k_gemm_wmma<<<(tiles + 3) / 4, 128, 0, stream>>>(x, W_in, b_in, h, NNODES, INDIM, HID);
    }

    for (int l = 0; l < NLAYERS; ++l) {
        int tiles = (NNODES / 16) * (HID / 16);
        k_gemm_wmma<<<(tiles + 3) / 4, 128, 0, stream>>>(h, Wl + (size_t)l * HID * HID,
                                                         bl + l * HID, xl, NNODES, HID, HID);
        k_gemm_wmma<<<(tiles + 3) / 4, 128, 0, stream>>>(h, Wr + (size_t)l * HID * HID,
                                                         br + l * HID, xr, NNODES, HID, HID);
        hipMemsetAsync(agg,  0,    (size_t)NNODES * HID   * sizeof(float), stream);
        hipMemsetAsync(den,  0,    (size_t)NNODES * HEADS * sizeof(float), stream);
        hipMemsetAsync(amax, 0x80, (size_t)NNODES * HEADS * sizeof(int),   stream); // ~ -3.4e38 ordered
        k_edge_alpha<<<((size_t)etot * 32 + 255) / 256, 256, 0, stream>>>(
            srcArr, dstArr, eatt, eaself, xl, xr, We + (size_t)l * HID, att + (size_t)l * HID,
            exbuf, amax);
        k_edge_exp<<<((size_t)etot * HEADS + 255) / 256, 256, 0, stream>>>(dstArr, exbuf, amax, den);
        k_edge_scatter<<<((size_t)etot * 32 + 255) / 256, 256, 0, stream>>>(
            srcArr, dstArr, exbuf, den, xl, agg);
        k_update_h<<<((size_t)NNODES * 32 + 255) / 256, 256, 0, stream>>>(
            h, agg, bc + l * HID, ln_g + l * HID, ln_b + l * HID);
    }

    // node_emb = h @ Wn + bn
    {
        int tiles = (NNODES / 16) * (EMB / 16);
        k_gemm_wmma<<<(tiles + 3) / 4, 128, 0, stream>>>(h, Wn, bn, node_emb, NNODES, HID, EMB);
    }

    // graph pooling + MLP
    hipMemsetAsync(gsum, 0, NB * HID * sizeof(float), stream);
    hipMemsetAsync(gcnt, 0, NB * sizeof(float), stream);
    k_batch_sum<<<((size_t)NNODES * 32 + 255) / 256, 256, 0, stream>>>(h, batch, gsum, gcnt);
    k_graph_mlp<<<1, 128, 0, stream>>>(gsum, gcnt, Wg1, bg1, Wg2, bg2, graph_emb);

    // third output: h
    hipMemcpyAsync(h_out, h, (size_t)NNODES * HID * sizeof(float),
                   hipMemcpyDeviceToDevice, stream);
}